// MLPmod_85701777424559
// MI455X (gfx1250) — compile-verified
//
#include <hip/hip_runtime.h>

typedef __attribute__((ext_vector_type(2))) float v2f;
typedef __attribute__((ext_vector_type(8))) float v8f;

// ---- kernel A tiling (featurizer + GEMM1) ----
#define TMA 128       // batch rows per block
#define TNA 128       // output cols per block
#define TKA 32        // k-chunk depth
#define LDAA 36       // As stride (36r mod 64 distinct -> conflict-free)
#define LDBA 264      // Bsp row stride: 128 cols * 2 + 8 pad (16B-aligned rows)

// ---- kernel B tiling (GEMM2 + GEMM3) ----
#define TM  128
#define TN  64
#define TK  64
#define LDA 68
#define LDB 68

// ---------------------------------------------------------------------------
// Kernel A: fused per-feature featurizer + GEMM1 + bias + ReLU
//   v[b,k] = relu(x[b, k%12] * Wf[k%12, k/12] + bf[k%12, k/12]),  K = 12288
//   out0[b,n] = relu( sum_k v[b,k] * W0[k,n] + b0[n] ),           N = 1024
// ---------------------------------------------------------------------------
__global__ __launch_bounds__(256)
void fused_feat_gemm1(const float* __restrict__ x,    // [B,12]
                      const float* __restrict__ Wf,   // [12,1024]
                      const float* __restrict__ bfp,  // [12,1024]
                      const float* __restrict__ W0,   // [12288,1024]
                      const float* __restrict__ b0,   // [1024]
                      float* __restrict__ out0)       // [B,1024]
{
    __shared__ float Xs[TMA][12];
    __shared__ float As[TMA][LDAA];
    __shared__ float Bsp[TKA / 2][LDBA];   // K-pair interleaved W0 tile

    const int tid = threadIdx.x;
    const int rb  = blockIdx.x * TMA;
    const int n0  = blockIdx.y * TNA;

    // preload x tile (128 x 12)
    for (int idx = tid; idx < TMA * 12; idx += 256) {
        int r = idx / 12, c = idx % 12;
        Xs[r][c] = x[(size_t)(rb + r) * 12 + c];
    }

    const int lane = tid & 31;
    const int half = lane >> 4;                 // 0 / 1
    const int lm   = lane & 15;
    const int wm   = ((tid >> 5) >> 1) * 32;    // wave row offset: 0,32,64,96
    const int wn   = ((tid >> 5) & 1) * 64;     // wave col offset: 0,64

    v8f zero = {};
    v8f acc[2][4];
    #pragma unroll
    for (int mt = 0; mt < 2; ++mt)
        #pragma unroll
        for (int nt = 0; nt < 4; ++nt)
            acc[mt][nt] = zero;

    __syncthreads();

    for (int kc = 0; kc < 12288; kc += TKA) {
        // --- build featurized A tile (128 x 32) in LDS ---
        #pragma unroll 4
        for (int s = 0; s < (TMA * TKA) / 256; ++s) {   // 16 elems/thread
            int idx = tid + s * 256;
            int kk  = idx & (TKA - 1);
            int r   = idx >> 5;
            int k   = kc + kk;
            int h   = k / 12;
            int i   = k - h * 12;
            float v = Xs[r][i] * Wf[i * 1024 + h] + bfp[i * 1024 + h];
            As[r][kk] = v > 0.0f ? v : 0.0f;
        }
        // --- W0 tile, K-pair interleaved: Bsp[kp][2*col+v] = W0[kc+2kp+v][n0+col] ---
        #pragma unroll
        for (int s = 0; s < (TKA / 2) * (TNA / 4) / 256; ++s) {  // 2 slots/thread
            int idx = tid + s * 256;
            int kp  = idx >> 5;          // 0..15
            int c4  = idx & 31;          // group of 4 columns
            const float4 w0 = *reinterpret_cast<const float4*>(
                &W0[(size_t)(kc + 2 * kp + 0) * 1024 + n0 + c4 * 4]);
            const float4 w1 = *reinterpret_cast<const float4*>(
                &W0[(size_t)(kc + 2 * kp + 1) * 1024 + n0 + c4 * 4]);
            float* dst = &Bsp[kp][c4 * 8];   // 16B-aligned, contiguous 8 floats
            dst[0] = w0.x; dst[1] = w1.x;
            dst[2] = w0.y; dst[3] = w1.y;
            dst[4] = w0.z; dst[5] = w1.z;
            dst[6] = w0.w; dst[7] = w1.w;
        }
        __syncthreads();

        // --- 8 k-steps, 8 WMMAs each (2x4 sixteen-tiles per wave) ---
        #pragma unroll
        for (int ks = 0; ks < TKA / 4; ++ks) {
            v2f a[2], b[4];
            #pragma unroll
            for (int mt = 0; mt < 2; ++mt)    // A: M = lane&15, K = 2*half + v
                a[mt] = *reinterpret_cast<const v2f*>(
                    &As[wm + mt * 16 + lm][ks * 4 + half * 2]);
            #pragma unroll
            for (int nt = 0; nt < 4; ++nt)    // B: N = lane&15, contiguous K pair
                b[nt] = *reinterpret_cast<const v2f*>(
                    &Bsp[ks * 2 + half][(wn + nt * 16 + lm) * 2]);
            #pragma unroll
            for (int mt = 0; mt < 2; ++mt)
                #pragma unroll
                for (int nt = 0; nt < 4; ++nt)
                    acc[mt][nt] = __builtin_amdgcn_wmma_f32_16x16x4_f32(
                        false, a[mt], false, b[nt], (short)0, acc[mt][nt], false, false);
        }
        __syncthreads();
    }

    // --- epilogue: bias + ReLU, store ---
    #pragma unroll
    for (int mt = 0; mt < 2; ++mt) {
        #pragma unroll
        for (int nt = 0; nt < 4; ++nt) {
            int col = n0 + wn + nt * 16 + lm;
            float bias = b0[col];
            #pragma unroll
            for (int r = 0; r < 8; ++r) {
                int row = rb + wm + mt * 16 + half * 8 + r;  // C: M = r + 8*half
                float v = acc[mt][nt][r] + bias;
                out0[(size_t)row * 1024 + col] = v > 0.0f ? v : 0.0f;
            }
        }
    }
}

// ---------------------------------------------------------------------------
// Kernel B: GEMM2 + bias + ReLU, fused with GEMM3 row-dot + bias
//   out1[b,n] = relu( sum_k out0[b,k] * W1[k,n] + b1[n] )   (K=1024, N=512)
//   out[b]    = sum_n out1[b,n] * W2[n] + b2
// ---------------------------------------------------------------------------
__global__ __launch_bounds__(256)
void fused_gemm23(const float* __restrict__ A0,   // [B,1024] (ws)
                  const float* __restrict__ W1,   // [1024,512]
                  const float* __restrict__ b1,   // [512]
                  const float* __restrict__ W2,   // [512]
                  const float* __restrict__ b2,   // [1]
                  float* __restrict__ out)        // [B]
{
    __shared__ float As[TM][LDA];
    __shared__ float Bs[TK][LDB];

    const int tid = threadIdx.x;
    const int rb  = blockIdx.x * TM;

    const int lane = tid & 31;
    const int half = lane >> 4;
    const int lm   = lane & 15;
    const int wm   = ((tid >> 5) >> 1) * 32;
    const int wn   = ((tid >> 5) & 1) * 32;

    float rowsum = 0.0f;   // meaningful for tid < 128 (row = tid)
    v8f zero = {};

    for (int nb = 0; nb < 512; nb += TN) {
        v8f acc[2][2];
        acc[0][0] = zero; acc[0][1] = zero; acc[1][0] = zero; acc[1][1] = zero;

        for (int kc = 0; kc < 1024; kc += TK) {
            #pragma unroll
            for (int s = 0; s < (TM * TK) / (4 * 256); ++s) {  // 8 float4/thread
                int idx = tid + s * 256;
                int r   = idx >> 4;
                int c4  = idx & 15;
                *reinterpret_cast<float4*>(&As[r][c4 * 4]) =
                    *reinterpret_cast<const float4*>(&A0[(size_t)(rb + r) * 1024 + kc + c4 * 4]);
            }
            #pragma unroll
            for (int s = 0; s < (TK * TN) / (4 * 256); ++s) {  // 4 float4/thread
                int idx = tid + s * 256;
                int r   = idx >> 4;
                int c4  = idx & 15;
                *reinterpret_cast<float4*>(&Bs[r][c4 * 4]) =
                    *reinterpret_cast<const float4*>(&W1[(size_t)(kc + r) * 512 + nb + c4 * 4]);
            }
            __syncthreads();

            #pragma unroll
            for (int ks = 0; ks < TK / 4; ++ks) {
                v2f a[2], b[2];
                #pragma unroll
                for (int mt = 0; mt < 2; ++mt) {
                    int row = wm + mt * 16 + lm;
                    a[mt][0] = As[row][ks * 4 + half * 2 + 0];
                    a[mt][1] = As[row][ks * 4 + half * 2 + 1];
                }
                #pragma unroll
                for (int nt = 0; nt < 2; ++nt) {
                    int col = wn + nt * 16 + lm;
                    b[nt][0] = Bs[ks * 4 + half * 2 + 0][col];
                    b[nt][1] = Bs[ks * 4 + half * 2 + 1][col];
                }
                #pragma unroll
                for (int mt = 0; mt < 2; ++mt)
                    #pragma unroll
                    for (int nt = 0; nt < 2; ++nt)
                        acc[mt][nt] = __builtin_amdgcn_wmma_f32_16x16x4_f32(
                            false, a[mt], false, b[nt], (short)0, acc[mt][nt], false, false);
            }
            __syncthreads();
        }

        // bias + ReLU into reused A-tile LDS (128 x 64 out1 chunk)
        #pragma unroll
        for (int mt = 0; mt < 2; ++mt) {
            #pragma unroll
            for (int nt = 0; nt < 2; ++nt) {
                int lcol = wn + nt * 16 + lm;
                float bias = b1[nb + lcol];
                #pragma unroll
                for (int r = 0; r < 8; ++r) {
                    int row = wm + mt * 16 + half * 8 + r;
                    float v = acc[mt][nt][r] + bias;
                    As[row][lcol] = v > 0.0f ? v : 0.0f;
                }
            }
        }
        __syncthreads();

        // GEMM3 partial: dot of this 64-wide chunk with W2
        if (tid < TM) {
            float s = 0.0f;
            #pragma unroll 8
            for (int c = 0; c < TN; ++c)
                s += As[tid][c] * W2[nb + c];
            rowsum += s;
        }
        __syncthreads();
    }

    if (tid < TM)
        out[rb + tid] = rowsum + b2[0];
}

// ---------------------------------------------------------------------------
extern "C" void kernel_launch(void* const* d_in, const int* in_sizes, int n_in,
                              void* d_out, int out_size, void* d_ws, size_t ws_size,
                              hipStream_t stream) {
    (void)in_sizes; (void)n_in; (void)out_size; (void)ws_size;
    const float* x  = (const float*)d_in[0];
    const float* Wf = (const float*)d_in[1];
    const float* bf = (const float*)d_in[2];
    const float* W0 = (const float*)d_in[3];
    const float* b0 = (const float*)d_in[4];
    const float* W1 = (const float*)d_in[5];
    const float* b1 = (const float*)d_in[6];
    const float* W2 = (const float*)d_in[7];
    const float* b2 = (const float*)d_in[8];

    float* out0 = (float*)d_ws;            // [16384,1024] fp32 = 64 MB scratch

    dim3 gridA(16384 / TMA, 1024 / TNA);   // 128 x 8 blocks
    hipLaunchKernelGGL(fused_feat_gemm1, gridA, dim3(256), 0, stream,
                       x, Wf, bf, W0, b0, out0);

    dim3 gridB(16384 / TM);                // 128 blocks
    hipLaunchKernelGGL(fused_gemm23, gridB, dim3(256), 0, stream,
                       out0, W1, b1, W2, b2, (float*)d_out);
}